// SimpleSAGE_7567732375847
// MI455X (gfx1250) — compile-verified
//
#include <hip/hip_runtime.h>
#include <hip/hip_bf16.h>

#define N_NODES 50000
#define N_EDGES 640000
#define DIM 128

typedef __attribute__((ext_vector_type(2))) float v2f;
typedef __attribute__((ext_vector_type(8))) float v8f;

// ---------------------------------------------------------------------------
// Zero out agg + cnt scratch (poisoned with 0xAA by the harness).
// ---------------------------------------------------------------------------
__global__ void sage_zero_kernel(float* __restrict__ p, int n) {
    int i = blockIdx.x * blockDim.x + threadIdx.x;
    if (i < n) p[i] = 0.0f;
}

// ---------------------------------------------------------------------------
// Edge scatter: 32 threads per edge; each thread moves 4 contiguous floats of
// x[src] into agg[dst] with global_atomic_add_f32. agg (25.6MB) sits in L2.
// ---------------------------------------------------------------------------
__global__ void sage_scatter_kernel(const float* __restrict__ x,
                                    const int* __restrict__ src,
                                    const int* __restrict__ dst,
                                    float* __restrict__ agg,
                                    float* __restrict__ cnt) {
    long long idx = (long long)blockIdx.x * blockDim.x + threadIdx.x;
    int e     = (int)(idx >> 5);
    int chunk = (int)(idx & 31);          // 32 chunks x 4 floats = 128
    if (e >= N_EDGES) return;
    int s = src[e];
    int d = dst[e];
    const float4 v = *reinterpret_cast<const float4*>(x + (long long)s * DIM + chunk * 4);
    float* a = agg + (long long)d * DIM + chunk * 4;
    atomicAdd(a + 0, v.x);
    atomicAdd(a + 1, v.y);
    atomicAdd(a + 2, v.z);
    atomicAdd(a + 3, v.w);
    if (chunk == 0) atomicAdd(cnt + d, 1.0f);
}

// ---------------------------------------------------------------------------
// Fused SAGE layer GEMM:
//   out[16 nodes x 128] = (agg/deg) @ wl^T + bias + xin @ wr^T   (+ ReLU)
// Block = 256 threads = 8 wave32s. Wave w owns output columns [16w, 16w+16).
// K=128 swept in steps of 4 with V_WMMA_F32_16X16X4_F32 (fp32 matrix core).
// ---------------------------------------------------------------------------
__global__ __launch_bounds__(256)
void sage_gemm_kernel(const float* __restrict__ agg,
                      const float* __restrict__ cnt,
                      const float* __restrict__ xin,
                      const float* __restrict__ wl,   // [128,128] row-major (out,in)
                      const float* __restrict__ bias, // [128]
                      const float* __restrict__ wr,   // [128,128]
                      float* __restrict__ out,
                      int applyRelu) {
    __shared__ float sAgg[16 * DIM];
    __shared__ float sX[16 * DIM];

    const int rowBase = blockIdx.x * 16;
    const int t = threadIdx.x;

    // Cooperative stage of the 16-row A panels into LDS; fold in 1/deg.
    for (int i = t; i < 16 * DIM; i += 256) {
        int r = i >> 7;                   // row within tile
        int f = i & 127;                  // feature
        int node = rowBase + r;
        float inv = 1.0f / fmaxf(cnt[node], 1.0f);
        sAgg[i] = agg[(long long)node * DIM + f] * inv;
        sX[i]   = xin[(long long)node * DIM + f];
    }
    __syncthreads();

    const int wave    = t >> 5;           // 0..7 -> column tile
    const int lane    = t & 31;
    const int colBase = wave * 16;
    const int mrow    = lane & 15;        // A: row M; B/C: column N
    const int khalf   = (lane >> 4) * 2;  // lanes 0-15 hold K {0,1}; 16-31 hold K {2,3}

    v8f c = {};                           // one 16x16 fp32 accumulator, both GEMMs chain into it
    for (int kk = 0; kk < DIM; kk += 4) {
        const int ka = kk + khalf;
        v2f aAgg, aX, bL, bR;
        // A fragments (16x4 fp32) from LDS
        aAgg.x = sAgg[mrow * DIM + ka];
        aAgg.y = sAgg[mrow * DIM + ka + 1];
        aX.x   = sX[mrow * DIM + ka];
        aX.y   = sX[mrow * DIM + ka + 1];
        // B fragments (4x16 fp32): B[k][n] = W[n][k]; contiguous 2-float loads, L2-resident
        const float* pL = wl + (long long)(colBase + mrow) * DIM + ka;
        const float* pR = wr + (long long)(colBase + mrow) * DIM + ka;
        bL.x = pL[0]; bL.y = pL[1];
        bR.x = pR[0]; bR.y = pR[1];
        // D = A*B + C  (8 args: neg_a, A, neg_b, B, c_mod, C, reuse_a, reuse_b)
        c = __builtin_amdgcn_wmma_f32_16x16x4_f32(false, aAgg, false, bL, (short)0, c, false, false);
        c = __builtin_amdgcn_wmma_f32_16x16x4_f32(false, aX,   false, bR, (short)0, c, false, false);
    }

    // C/D layout: VGPR v -> M = v + (lane>=16 ? 8 : 0), N = lane&15
    const float bv = bias[colBase + mrow];
    const int rowOff = (lane >> 4) * 8;
#pragma unroll
    for (int v = 0; v < 8; ++v) {
        float val = c[v] + bv;
        if (applyRelu) val = fmaxf(val, 0.0f);
        out[(long long)(rowBase + rowOff + v) * DIM + colBase + mrow] = val;
    }
}

// ---------------------------------------------------------------------------
// Launch: two SAGE layers. ws layout: agg[N*128] | cnt[N] | h[N*128]
// ---------------------------------------------------------------------------
extern "C" void kernel_launch(void* const* d_in, const int* in_sizes, int n_in,
                              void* d_out, int out_size, void* d_ws, size_t ws_size,
                              hipStream_t stream) {
    const float* x   = (const float*)d_in[0];
    const int*   ei  = (const int*)d_in[1];   // [2, E] flat: src row then dst row
    const float* w1l = (const float*)d_in[2];
    const float* b1  = (const float*)d_in[3];
    const float* w1r = (const float*)d_in[4];
    const float* w2l = (const float*)d_in[5];
    const float* b2  = (const float*)d_in[6];
    const float* w2r = (const float*)d_in[7];
    float* out = (float*)d_out;

    float* agg = (float*)d_ws;
    float* cnt = agg + (size_t)N_NODES * DIM;
    float* h   = cnt + N_NODES;

    const int* src = ei;
    const int* dst = ei + N_EDGES;

    const int zeroN      = N_NODES * DIM + N_NODES;       // agg + cnt contiguous
    const int zeroBlocks = (zeroN + 255) / 256;
    const long long scatterThreads = (long long)N_EDGES * 32;
    const int scatterBlocks = (int)((scatterThreads + 255) / 256);
    const int gemmBlocks = N_NODES / 16;                  // 50000 % 16 == 0

    // Layer 1: h = relu(sage_conv(x))
    sage_zero_kernel<<<zeroBlocks, 256, 0, stream>>>(agg, zeroN);
    sage_scatter_kernel<<<scatterBlocks, 256, 0, stream>>>(x, src, dst, agg, cnt);
    sage_gemm_kernel<<<gemmBlocks, 256, 0, stream>>>(agg, cnt, x, w1l, b1, w1r, h, 1);

    // Layer 2: out = sage_conv(h)
    sage_zero_kernel<<<zeroBlocks, 256, 0, stream>>>(agg, zeroN);
    sage_scatter_kernel<<<scatterBlocks, 256, 0, stream>>>(h, src, dst, agg, cnt);
    sage_gemm_kernel<<<gemmBlocks, 256, 0, stream>>>(agg, cnt, h, w2l, b2, w2r, out, 0);
}